// TorchNet_81028853006841
// MI455X (gfx1250) — compile-verified
//
#include <hip/hip_runtime.h>
#include <hip/hip_bf16.h>
#include <math.h>

// ---------------------------------------------------------------------------
// Sparse child-sum network for MI455X (gfx1250, wave32).
//   valsT column-major: valsT[col*BATCH + b], 72 MB -> resident in 192 MB L2.
//   Each lane owns 4 batch rows (float4) -> every gather is global_load_b128
//   in saddr form (scalar column base, loop-invariant VGPR offset).
//   Indices are block-uniform -> SMEM s_load + SALU address math.
//   Input transpose stages X into LDS with the CDNA5 async copy engine
//   (global_load_async_to_lds_b32 / s_wait_asynccnt).
// ---------------------------------------------------------------------------

#define BATCH    1024
#define N_IN     1024
#define DEG      32
#define NPB      2          // neurons per block
#define NLAYERS  9

typedef float v4f __attribute__((ext_vector_type(4)));

#if __has_builtin(__builtin_amdgcn_tanhf)
#define FAST_TANH(x) __builtin_amdgcn_tanhf(x)
#else
#define FAST_TANH(x) tanhf(x)
#endif

#define USE_ASYNC_LDS 1

// --- X (BATCH x N_IN row-major) -> V (col-major: V[c*BATCH + b]) ------------
__global__ __launch_bounds__(256) void xpose_kernel(const float* __restrict__ X,
                                                    float* __restrict__ V) {
  __shared__ float tile[32][33];
  const int tx = threadIdx.x, ty = threadIdx.y;
  const int c0 = blockIdx.x * 32;
  const int b0 = blockIdx.y * 32;

#if USE_ASYNC_LDS
  // stage the 32x32 tile global -> LDS via the async copy engine
#pragma unroll
  for (int i = 0; i < 32; i += 8) {
    const float* g = X + (size_t)(b0 + ty + i) * N_IN + (c0 + tx);
    unsigned lo = (unsigned)(unsigned long long)&tile[ty + i][tx];
    asm volatile("global_load_async_to_lds_b32 %0, %1, off"
                 :: "v"(lo), "v"(g) : "memory");
  }
  asm volatile("s_wait_asynccnt 0x0" ::: "memory");
#else
#pragma unroll
  for (int i = 0; i < 32; i += 8)
    tile[ty + i][tx] = X[(size_t)(b0 + ty + i) * N_IN + (c0 + tx)];
#endif
  __syncthreads();
#pragma unroll
  for (int i = 0; i < 32; i += 8)
    V[(size_t)(c0 + ty + i) * BATCH + (b0 + tx)] = tile[tx][ty + i];
}

// --- one layer: each block computes NPB neurons over the full batch ---------
__global__ __launch_bounds__(256) void layer_kernel(float* __restrict__ V,
                                                    const int* __restrict__ cidx,
                                                    const float* __restrict__ wptr,
                                                    int ptr, int offset, int last,
                                                    float* __restrict__ out) {
  const int tid   = threadIdx.x;
  const int jbase = blockIdx.x * NPB;
  const float w   = wptr[0];               // uniform -> s_load
  const int b     = tid * 4;               // 256 threads * 4 rows = full batch

#pragma unroll 1
  for (int n = 0; n < NPB; ++n) {
    const int j = jbase + n;
    const size_t ibase = (size_t)(ptr + j) * DEG;

    // block-uniform index row: uniform address -> SMEM (merged s_load_bN)
    int cols[DEG];
#pragma unroll
    for (int k = 0; k < DEG; ++k)
      cols[k] = cidx[ibase + k];

    v4f acc = (v4f)0.0f;
#pragma unroll
    for (int k = 0; k < DEG; ++k) {
      // force SGPR so the gather selects saddr form (SALU base, fixed voffset)
      const int c = __builtin_amdgcn_readfirstlane(cols[k]);
      const v4f* col = (const v4f*)(V + (size_t)c * BATCH);
      acc += col[tid];                     // global_load_b128 + v_pk_add_f32
    }
    acc *= w;
    const float r0 = FAST_TANH(acc.x);
    const float r1 = FAST_TANH(acc.y);
    const float r2 = FAST_TANH(acc.z);
    const float r3 = FAST_TANH(acc.w);

    if (!last) {
      v4f r; r.x = r0; r.y = r1; r.z = r2; r.w = r3;
      *(v4f*)(V + (size_t)(offset + j) * BATCH + b) = r;   // coalesced b128
    } else {
      // final layer: row-major output out[b][j]; 4 MB, combined in L2
      out[(size_t)(b + 0) * 1024 + j] = r0;
      out[(size_t)(b + 1) * 1024 + j] = r1;
      out[(size_t)(b + 2) * 1024 + j] = r2;
      out[(size_t)(b + 3) * 1024 + j] = r3;
    }
  }
}

extern "C" void kernel_launch(void* const* d_in, const int* in_sizes, int n_in,
                              void* d_out, int out_size, void* d_ws, size_t ws_size,
                              hipStream_t stream) {
  const float* X    = (const float*)d_in[0];
  const float* w    = (const float*)d_in[1];   // scalar weight on device
  const int*   cidx = (const int*)d_in[2];
  float* out = (float*)d_out;
  float* V   = (float*)d_ws;                   // 18432*1024 fp32 = 72 MB

  static const int LAYER_SZ[NLAYERS] = {2048,2048,2048,2048,2048,2048,2048,2048,1024};

  dim3 tb(32, 8), tg(N_IN / 32, BATCH / 32);
  hipLaunchKernelGGL(xpose_kernel, tg, tb, 0, stream, X, V);

  int ptr = 0, offset = N_IN;
  for (int l = 0; l < NLAYERS; ++l) {
    const int sz   = LAYER_SZ[l];
    const int last = (l == NLAYERS - 1);
    hipLaunchKernelGGL(layer_kernel, dim3(sz / NPB), dim3(256), 0, stream,
                       V, cidx, w, ptr, offset, last, out);
    ptr += sz;
    offset += sz;
  }
}